// QEncoder_52793738003017
// MI455X (gfx1250) — compile-verified
//
#include <hip/hip_runtime.h>
#include <hip/hip_bf16.h>
#include <math.h>

// CDNA5 / gfx1250. wave32 only. Uses V_WMMA_F32_16X16X4_F32 (fp32 matrix core).
typedef float v2f __attribute__((ext_vector_type(2)));
typedef float v8f __attribute__((ext_vector_type(8)));

#define DIM    64
#define NB     16      // batch elements (columns) per block == per wave
#define LSTR   17      // LDS row stride (pad to avoid bank conflicts)
#define MTERMS 10      // Taylor order per segment (|z| <= 0.6 -> err ~ 1e-10)

// K = a^dag - a (real antisymmetric tridiagonal):
//   K[i, i-1] = sqrt(i),  K[i, i+1] = -sqrt(i+1)
__device__ __forceinline__ float kval(int i, int j) {
  if (j == i - 1) return sqrtf((float)i);
  if (j == i + 1) return -sqrtf((float)(i + 1));
  return 0.0f;
}

__global__ __launch_bounds__(32)
void qenc_rho_kernel(const float* __restrict__ inputs, float* __restrict__ out) {
  __shared__ float Tl[DIM * LSTR];    // term block / psi block, 64 x 16 (stride 17)
  __shared__ float s_lds[DIM];        // squeezed vacuum vector
  __shared__ float alpha_lds[NB];
  __shared__ float invn_lds[NB];
  __shared__ int   nseg_lds;

  const int lane = threadIdx.x;   // 0..31, one wave per block
  const int half = lane >> 4;     // lane-half for WMMA operand layouts
  const int m    = lane & 15;     // column / M-row index within tiles
  const int b0   = blockIdx.x * NB;

  // ---- alphas + squeezed vacuum state (closed-form recurrence) -----------
  if (lane < NB) alpha_lds[lane] = 0.5f * inputs[b0 + lane];
  if (lane == 0) {
    const float r  = 0.5f;
    const float th = tanhf(r);
    float c = rsqrtf(coshf(r));           // c0 = 1/sqrt(cosh r)
    for (int i = 0; i < DIM; ++i) s_lds[i] = 0.0f;
    s_lds[0] = c;
    for (int n = 0; 2 * n + 2 < DIM; ++n) {
      c *= -th * sqrtf((float)(2 * n + 1) / (float)(2 * n + 2));
      s_lds[2 * n + 2] = c;               // odd amplitudes are zero
    }
  }
  __syncthreads();
  if (lane == 0) {
    float amax = 0.0f;
    for (int b = 0; b < NB; ++b) amax = fmaxf(amax, fabsf(alpha_lds[b]));
    // ||K||_2 <= 2*sqrt(63) ~= 15.87 ; keep per-segment norm <= 0.6
    int ns = (int)ceilf(amax * 26.5f);
    nseg_lds = ns < 1 ? 1 : (ns > 4096 ? 4096 : ns);
  }
  __syncthreads();
  const int nseg = nseg_lds;   // uniform across wave; deterministic in inputs

  // ---- precompute A-operand tiles of K (16x4 f32 layout, 2 VGPRs) --------
  // A tile (rt, t): rows 16rt..16rt+15, cols 4t..4t+3. Nonzero only for
  // t in [4rt-1, 4rt+4] (22 tiles total out of 64).
  float Ax[4][6], Ay[4][6];
#pragma unroll
  for (int rt = 0; rt < 4; ++rt) {
#pragma unroll
    for (int u = 0; u < 6; ++u) {
      int t = 4 * rt - 1 + u;
      float ax = 0.0f, ay = 0.0f;
      if (t >= 0 && t <= 15) {
        int i  = 16 * rt + m;           // global row (M = lane&15)
        int j0 = 4 * t + 2 * half;      // lanes 0-15 hold K-cols {0,1}, 16-31 {2,3}
        ax = kval(i, j0);
        ay = kval(i, j0 + 1);
      }
      Ax[rt][u] = ax;
      Ay[rt][u] = ay;
    }
  }

  // ---- V (64x16, C/D layout: 4 tiles of v8f): init every column to s -----
  v8f V[4], T[4];
#pragma unroll
  for (int rt = 0; rt < 4; ++rt)
#pragma unroll
    for (int p = 0; p < 8; ++p)
      V[rt][p] = s_lds[16 * rt + p + 8 * half];

  const float cl = alpha_lds[m] / (float)nseg;  // per-column scaled step

  // ---- psi = exp(alpha K) s, segmented Taylor; K@T via f32 WMMA ----------
  for (int seg = 0; seg < nseg; ++seg) {
#pragma unroll
    for (int rt = 0; rt < 4; ++rt) T[rt] = V[rt];
    for (int j = 1; j <= MTERMS; ++j) {
      // publish T (C-layout regs) to LDS so it can be re-read in B-layout
#pragma unroll
      for (int rt = 0; rt < 4; ++rt)
#pragma unroll
        for (int p = 0; p < 8; ++p)
          Tl[(16 * rt + p + 8 * half) * LSTR + m] = T[rt][p];
      __syncthreads();
      const float f = cl / (float)j;
#pragma unroll
      for (int rt = 0; rt < 4; ++rt) {
        v8f acc = {};
#pragma unroll
        for (int u = 0; u < 6; ++u) {
          int t = 4 * rt - 1 + u;
          if (t < 0 || t > 15) continue;     // compile-time folded
          v2f a, b;
          a.x = Ax[rt][u];
          a.y = Ay[rt][u];
          // B 4x16: VGPR v, lanes<16 -> K-row 4t+v ; lanes>=16 -> 4t+v+2
          int row = 4 * t + 2 * half;
          b.x = Tl[row * LSTR + m];
          b.y = Tl[(row + 1) * LSTR + m];
          acc = __builtin_amdgcn_wmma_f32_16x16x4_f32(
              false, a, false, b, (short)0, acc, false, false);
        }
        T[rt] = acc * f;                     // per-column (per-lane) scale
      }
      __syncthreads();                       // reads done before next publish
#pragma unroll
      for (int rt = 0; rt < 4; ++rt) V[rt] += T[rt];
    }
  }

  // ---- psi -> LDS, per-column normalization ------------------------------
#pragma unroll
  for (int rt = 0; rt < 4; ++rt)
#pragma unroll
    for (int p = 0; p < 8; ++p)
      Tl[(16 * rt + p + 8 * half) * LSTR + m] = V[rt][p];
  __syncthreads();
  if (lane < NB) {
    float s2 = 0.0f;
    for (int i = 0; i < DIM; ++i) {
      float v = Tl[i * LSTR + lane];
      s2 += v * v;
    }
    invn_lds[lane] = rsqrtf(s2);
  }
  __syncthreads();

  // ---- rho = psi psi^T as complex64 (imag = 0); store-bandwidth bound ----
  // Each lane stores one float4 = two complex elements per row of rho.
  for (int b = 0; b < NB; ++b) {
    const float inv = invn_lds[b];
    const float pj0 = Tl[(2 * lane) * LSTR + b] * inv;
    const float pj1 = Tl[(2 * lane + 1) * LSTR + b] * inv;
    float* dst = out + (size_t)(b0 + b) * (DIM * DIM * 2);
    for (int i = 0; i < DIM; ++i) {
      float pi = Tl[i * LSTR + b] * inv;    // LDS broadcast read
      float4 vv = make_float4(pi * pj0, 0.0f, pi * pj1, 0.0f);
      *(float4*)(dst + (size_t)i * (DIM * 2) + lane * 4) = vv;
    }
  }
}

extern "C" void kernel_launch(void* const* d_in, const int* in_sizes, int n_in,
                              void* d_out, int out_size, void* d_ws, size_t ws_size,
                              hipStream_t stream) {
  (void)n_in; (void)out_size; (void)d_ws; (void)ws_size;
  const float* inputs = (const float*)d_in[0];   // (4096,) float32
  // d_in[1] = vacuum state |0>; implicit in the closed-form squeezed vector.
  float* out = (float*)d_out;                    // (4096,64,64) complex64 -> float pairs
  const int batch  = in_sizes[0];
  const int blocks = batch / NB;                 // 4096/16 = 256 blocks, 1 wave each
  qenc_rho_kernel<<<blocks, 32, 0, stream>>>(inputs, out);
}